// DiffusionBlock_35399120454152
// MI455X (gfx1250) — compile-verified
//
#include <hip/hip_runtime.h>
#include <stdint.h>

// ---------------------------------------------------------------------------
// Anisotropic diffusion step, fused single pass, vectorized.
//
// Roofline: ~380 MB moved per launch vs ~1.3 GFLOP -> HBM-bandwidth bound
// (~16us floor @ 23.3 TB/s). No matmul structure -> WMMA not applicable.
// CDNA5 paths: GLOBAL_LOAD_ASYNC_TO_LDS_B128/B32 (ASYNCcnt) staging of the
// reused u tile, s_wait_asynccnt split-counter sync. float4 everywhere else
// to stay under the wave32 issue-rate budget.
// ---------------------------------------------------------------------------

#define BH 768
#define BW 768
#define WP 770               // a/b/c padded width (H+2 = W+2)
#define TH 16                // output tile height
#define TW 64                // output tile width
#define NTHREADS 256
#define SU_ST 72             // LDS row stride (floats); 288B = 16B-aligned rows
#define SV_ST 72

typedef float f4v __attribute__((ext_vector_type(4)));               // 16B aligned
typedef float f4u __attribute__((ext_vector_type(4), aligned(4)));   // 4B aligned

__global__ __launch_bounds__(NTHREADS)
void diffusion_step_kernel(const float* __restrict__ u,
                           const float* __restrict__ A,
                           const float* __restrict__ Bm,
                           const float* __restrict__ Cm,
                           const float* __restrict__ alpha_p,
                           const float* __restrict__ tau_p,
                           float* __restrict__ out)
{
    __shared__ __attribute__((aligned(16))) float su [18 * SU_ST];  // u tile, rows 0..17, cols 0..67
    __shared__ __attribute__((aligned(16))) float sv1[17 * SV_ST];  // flux fields, rows 0..16, cols 0..64
    __shared__ __attribute__((aligned(16))) float sv2[17 * SV_ST];
    __shared__ __attribute__((aligned(16))) float sv3[17 * SV_ST];
    __shared__ __attribute__((aligned(16))) float sv4[17 * SV_ST];

    const int plane = blockIdx.z;
    const int oi    = blockIdx.y * TH;
    const int oj    = blockIdx.x * TW;
    const int tid   = threadIdx.x;

    const float alpha = alpha_p[0];
    const float tau   = tau_p[0];

    const float* up = u  + (size_t)plane * BH * BW;
    const float* Ap = A  + (size_t)plane * WP * WP;
    const float* Bp = Bm + (size_t)plane * WP * WP;
    const float* Cp = Cm + (size_t)plane * WP * WP;

    // ---- Phase 1: async global->LDS staging of the u tile -----------------
    const bool interior = (blockIdx.x != gridDim.x - 1) && (blockIdx.y != gridDim.y - 1);
    if (interior) {
        // 18 rows x 17 b128 chunks (68 cols), no clamping needed.
        for (int idx = tid; idx < 18 * 17; idx += NTHREADS) {
            const int r = idx / 17;
            const int q = idx - r * 17;
            const unsigned long long gaddr =
                (unsigned long long)(up + (size_t)(oi + r) * BW + (oj + q * 4));
            const unsigned lds_off = (unsigned)(uintptr_t)(&su[r * SU_ST + q * 4]);
            asm volatile("global_load_async_to_lds_b128 %0, %1, off"
                         :: "v"(lds_off), "v"(gaddr) : "memory");
        }
    } else {
        // Edge-replicated (clamped) scalar path: 18 rows x 66 cols.
        for (int idx = tid; idx < 18 * 66; idx += NTHREADS) {
            const int r = idx / 66;
            const int c = idx - r * 66;
            int gy = oi + r; if (gy > BH - 1) gy = BH - 1;
            int gx = oj + c; if (gx > BW - 1) gx = BW - 1;
            const unsigned long long gaddr =
                (unsigned long long)(up + (size_t)gy * BW + gx);
            const unsigned lds_off = (unsigned)(uintptr_t)(&su[r * SU_ST + c]);
            asm volatile("global_load_async_to_lds_b32 %0, %1, off"
                         :: "v"(lds_off), "v"(gaddr) : "memory");
        }
    }
#if __has_builtin(__builtin_amdgcn_s_wait_asynccnt)
    __builtin_amdgcn_s_wait_asynccnt(0);
#else
    asm volatile("s_wait_asynccnt 0" ::: "memory");
#endif
    __syncthreads();

    // ---- Phase 2: flux fields v1..v4 on 17 x 65 halo grid -----------------
    const float c1a  = (1.0f - alpha) * 0.5f;
    const float c2a  = alpha * 0.5f;
    const float bdev = 1.0f - 2.0f * alpha;

    // Vector part: 17 rows x 16 float4 chunks (cols 0..63).
    for (int idx = tid; idx < 17 * 16; idx += NTHREADS) {
        const int r  = idx >> 4;
        const int c0 = (idx & 15) * 4;
        const size_t o = (size_t)(oi + r + 1) * WP + (size_t)(oj + c0 + 1);
        const f4u a4 = *(const f4u*)(Ap + o);
        const f4u b4 = *(const f4u*)(Bp + o);
        const f4u g4 = *(const f4u*)(Cp + o);

        const f4v s0 = *(const f4v*)(&su[r * SU_ST + c0]);
        const float s0e = su[r * SU_ST + c0 + 4];
        const f4v s1 = *(const f4v*)(&su[(r + 1) * SU_ST + c0]);
        const float s1e = su[(r + 1) * SU_ST + c0 + 4];

        f4v r1, r2, r3, r4;
        #pragma unroll
        for (int k = 0; k < 4; ++k) {
            const float u00 = s0[k];
            const float u01 = (k < 3) ? s0[k + 1] : s0e;
            const float u10 = s1[k];
            const float u11 = (k < 3) ? s1[k + 1] : s1e;
            const float ux1 = u01 - u00;
            const float ux2 = u11 - u10;
            const float uy1 = u10 - u00;
            const float uy2 = u11 - u01;

            const float aw = a4[k], bw = b4[k], cw = g4[k];
            const float sgn  = (bw > 0.0f) ? 1.0f : ((bw < 0.0f) ? -1.0f : 0.0f);
            const float beta = bdev * sgn;
            const float w1a = aw * c1a, w2a = aw * c2a;
            const float w1b = bw * (1.0f - beta) * 0.25f;
            const float w2b = bw * (1.0f + beta) * 0.25f;
            const float w1c = cw * c1a, w2c = cw * c2a;

            r1[k] = w1a * ux1 + w2a * ux2 + w1b * uy1 + w2b * uy2;
            r2[k] = w2a * ux1 + w1a * ux2 + w2b * uy1 + w1b * uy2;
            r3[k] = w1b * ux1 + w2b * ux2 + w1c * uy1 + w2c * uy2;
            r4[k] = w2b * ux1 + w1b * ux2 + w2c * uy1 + w1c * uy2;
        }
        const int vo = r * SV_ST + c0;
        *(f4v*)(&sv1[vo]) = r1;
        *(f4v*)(&sv2[vo]) = r2;
        *(f4v*)(&sv3[vo]) = r3;
        *(f4v*)(&sv4[vo]) = r4;
    }
    // Scalar cleanup: column 64 (17 points).
    if (tid < 17) {
        const int r = tid, c = 64;
        const size_t o = (size_t)(oi + r + 1) * WP + (size_t)(oj + c + 1);
        const float aw = Ap[o], bw = Bp[o], cw = Cp[o];
        const float sgn  = (bw > 0.0f) ? 1.0f : ((bw < 0.0f) ? -1.0f : 0.0f);
        const float beta = bdev * sgn;
        const float w1a = aw * c1a, w2a = aw * c2a;
        const float w1b = bw * (1.0f - beta) * 0.25f;
        const float w2b = bw * (1.0f + beta) * 0.25f;
        const float w1c = cw * c1a, w2c = cw * c2a;
        const float u00 = su[ r      * SU_ST + c    ];
        const float u01 = su[ r      * SU_ST + c + 1];
        const float u10 = su[(r + 1) * SU_ST + c    ];
        const float u11 = su[(r + 1) * SU_ST + c + 1];
        const float ux1 = u01 - u00, ux2 = u11 - u10;
        const float uy1 = u10 - u00, uy2 = u11 - u01;
        const int vo = r * SV_ST + c;
        sv1[vo] = w1a * ux1 + w2a * ux2 + w1b * uy1 + w2b * uy2;
        sv2[vo] = w2a * ux1 + w1a * ux2 + w2b * uy1 + w1b * uy2;
        sv3[vo] = w1b * ux1 + w2b * ux2 + w1c * uy1 + w2c * uy2;
        sv4[vo] = w2b * ux1 + w1b * ux2 + w2c * uy1 + w1c * uy2;
    }
    __syncthreads();

    // ---- Phase 3: divergence + update, one float4 per thread --------------
    {
        float* op = out + (size_t)plane * BH * BW;
        const float ht = 0.5f * tau;
        const int r  = tid >> 4;           // 0..15
        const int c0 = (tid & 15) * 4;     // 0..60

        const f4v A1  = *(const f4v*)(&sv1[(r + 1) * SV_ST + c0]);
        const float a1e = sv1[(r + 1) * SV_ST + c0 + 4];
        const f4v B2  = *(const f4v*)(&sv2[ r      * SV_ST + c0]);
        const float b2e = sv2[ r      * SV_ST + c0 + 4];
        const f4v C3a = *(const f4v*)(&sv3[ r      * SV_ST + c0]);
        const float c3ae = sv3[ r      * SV_ST + c0 + 4];
        const f4v C3b = *(const f4v*)(&sv3[(r + 1) * SV_ST + c0]);
        const float c3be = sv3[(r + 1) * SV_ST + c0 + 4];
        const f4v D4a = *(const f4v*)(&sv4[ r      * SV_ST + c0]);
        const f4v D4b = *(const f4v*)(&sv4[(r + 1) * SV_ST + c0]);
        const f4v uc  = *(const f4v*)(&su [ r      * SU_ST + c0]);

        const f4v A1s  = { A1[1],  A1[2],  A1[3],  a1e  };  // v1 row r+1, col+1
        const f4v B2s  = { B2[1],  B2[2],  B2[3],  b2e  };  // v2 row r,   col+1
        const f4v C3as = { C3a[1], C3a[2], C3a[3], c3ae };  // v3 row r,   col+1
        const f4v C3bs = { C3b[1], C3b[2], C3b[3], c3be };  // v3 row r+1, col+1

        f4v res;
        #pragma unroll
        for (int k = 0; k < 4; ++k) {
            const float Au = (A1s[k] - A1[k]) + (B2s[k] - B2[k])
                           + (C3bs[k] - C3as[k]) + (D4b[k] - D4a[k]);
            res[k] = uc[k] + ht * Au;
        }
        *(f4v*)(op + (size_t)(oi + r) * BW + (oj + c0)) = res;
    }
}

extern "C" void kernel_launch(void* const* d_in, const int* in_sizes, int n_in,
                              void* d_out, int out_size, void* d_ws, size_t ws_size,
                              hipStream_t stream) {
    // Inputs: 0:u 1:a 2:b 3:c 4..7:grad stencils (fixed, folded in) 8:alpha 9:tau
    const float* u     = (const float*)d_in[0];
    const float* a     = (const float*)d_in[1];
    const float* b     = (const float*)d_in[2];
    const float* c     = (const float*)d_in[3];
    const float* alpha = (const float*)d_in[8];
    const float* tau   = (const float*)d_in[9];
    float* out = (float*)d_out;

    const int planes = in_sizes[0] / (BH * BW);   // B*C = 32
    dim3 grid(BW / TW, BH / TH, planes);          // (12, 48, 32)
    diffusion_step_kernel<<<grid, NTHREADS, 0, stream>>>(u, a, b, c, alpha, tau, out);
    (void)d_ws; (void)ws_size; (void)n_in; (void)out_size;
}